// ExpanderConv2d_1580547965660
// MI455X (gfx1250) — compile-verified
//
#include <hip/hip_runtime.h>

typedef float v2f __attribute__((ext_vector_type(2)));
typedef float v4f __attribute__((ext_vector_type(4)));
typedef float v8f __attribute__((ext_vector_type(8)));

#define CIN     256
#define COUT    256
#define HH      56
#define WW      56
#define SPATIAL (HH * WW)            // 3136
#define NPIX    (32 * SPATIAL)       // 100352
#define KTOT    (CIN * 9)            // 2304

#define TILE_M  128
#define TILE_N  128
#define KB      16                   // K slice per LDS stage
#define KP      (KB / 2)             // 8 K-pair rows per panel
#define NCHUNK  (KTOT / KB)          // 144
#define ROWSTRIDE (2 * TILE_M + 32)  // 288 floats: pair-interleave + bank skew
#define PANEL   (KP * ROWSTRIDE)     // 2304 floats per panel (9 KB)

// ---- CDNA5 async global->LDS path (ASYNCcnt), with sync fallback ----------
#if __has_builtin(__builtin_amdgcn_global_load_async_to_lds_b128)
#define HAVE_ASYNC 1
#else
#define HAVE_ASYNC 0
#endif

// Builtin signature (from hipcc diagnostic): generic-vector int4 pointers,
// pointee in AS1 (global) for src and AS3 (LDS) for dst, imm offset, imm cpol.
typedef int gv4i __attribute__((vector_size(16)));
typedef __attribute__((address_space(1))) gv4i* gas1_v4i_p;
typedef __attribute__((address_space(3))) gv4i* gas3_v4i_p;

__device__ __forceinline__ void async_copy16(const float* gsrc, float* ldst) {
#if HAVE_ASYNC
    __builtin_amdgcn_global_load_async_to_lds_b128(
        (gas1_v4i_p)(void*)gsrc, (gas3_v4i_p)(void*)ldst, 0, 0);
#else
    *(v4f*)ldst = *(const v4f*)gsrc;
#endif
}

__device__ __forceinline__ void wait_async() {
#if HAVE_ASYNC
#if __has_builtin(__builtin_amdgcn_s_wait_asynccnt)
    __builtin_amdgcn_s_wait_asynccnt(0);
#else
    asm volatile("s_wait_asynccnt 0" ::: "memory");
#endif
#endif
}

// ---------------------------------------------------------------------------
// Stage 1: fold mask into weights, pack to the WMMA-pair-interleaved layout
// g(k,cout) = (k>>1)*512 + cout*2 + (k&1), k = (kh*3+kw)*256 + cin.
// A-panel staging then becomes a contiguous 16B-granular async copy.
// ---------------------------------------------------------------------------
__global__ __launch_bounds__(256)
void prep_weights(const float* __restrict__ w,
                  const float* __restrict__ msk,
                  float* __restrict__ wt) {
    int g = blockIdx.x * 256 + threadIdx.x;
    if (g >= KTOT * COUT) return;
    int kp   = g >> 9;
    int rem  = g & 511;
    int cout = rem >> 1;
    int b    = rem & 1;
    int k    = (kp << 1) | b;
    int r    = k >> 8;                 // 0..8 tap
    int cin  = k & 255;
    size_t src = (size_t)(cout * CIN + cin) * 9 + r;  // OIHW flat
    wt[g] = w[src] * msk[src];
}

// ---------------------------------------------------------------------------
// Stage 2: double-buffered implicit GEMM, V_WMMA_F32_16X16X4_F32.
// 256 threads = 8 waves; 128x128 C-tile; wave (wm 0..3, wn 0..1) owns 32x64
// as 2x4 WMMA subtiles (8 v8f accumulators).
// ---------------------------------------------------------------------------
__global__ __launch_bounds__(256)
void conv_wmma(const float* __restrict__ x,
               const float* __restrict__ wt,
               float* __restrict__ out) {
    __shared__ __align__(16) float As[2][PANEL];
    __shared__ __align__(16) float Bs[2][PANEL];

    const int tid  = threadIdx.x;
    const int lane = tid & 31;
    const int wave = tid >> 5;
    const int wm   = wave >> 1;        // 0..3 : 32-row slab
    const int wn   = wave & 1;         // 0..1 : 64-col slab
    const int m0   = blockIdx.x * TILE_M;
    const int n0   = blockIdx.y * TILE_N;

    const int row   = lane & 15;       // f32 frag: M/N index within subtile
    const int koff2 = lane >> 4;       // 0/1 : which K-pair this half-wave reads

    // ---- per-thread staging geometry (loop-invariant) ----
    // B: thread covers (kk = kkbase + 2j, nn) for j=0..7
    const int nn     = tid & 127;
    const int kkbase = tid >> 7;       // 0 or 1
    const int p   = n0 + nn;           // global pixel (< NPIX by exact grid)
    const int ni  = p / SPATIAL;
    const int s0  = p - ni * SPATIAL;
    const int h   = s0 / WW;
    const int w   = s0 - h * WW;
    const int sb  = nn * 2 + kkbase;   // Bs store offset within pair-row
    // A: thread covers (kpl = akpl + 4it, inner) 16B units
    const int ainner = tid & 63;
    const int akpl   = tid >> 6;       // 0..3

    v8f acc[2][4] = {};                // [m-subtile][n-subtile]

    auto stageA = [&](int c, int d) {  // async DMA: weights panel -> LDS
        const int k0h = (c * KB) >> 1;
        #pragma unroll
        for (int it = 0; it < 2; ++it) {
            int kpl = akpl + 4 * it;
            const float* gsrc =
                wt + (size_t)(k0h + kpl) * (2 * COUT) + m0 * 2 + ainner * 4;
            float* ldst = &As[d][kpl * ROWSTRIDE + ainner * 4];
            async_copy16(gsrc, ldst);
        }
    };

    auto loadB = [&](int c, float* pre) {  // issue 8 global loads -> regs
        const int r    = c >> 4;           // tap (constant over chunk)
        const int kh   = r / 3;
        const int kw   = r - kh * 3;
        const int cinb = (c & 15) * KB;
        const int hs = h + kh - 1;
        const int ws = w + kw - 1;
        const bool ok = (hs >= 0) && (hs < HH) && (ws >= 0) && (ws < WW);
        const float* base =
            x + (((size_t)ni * CIN + cinb + kkbase) * HH + hs) * WW + ws;
        #pragma unroll
        for (int j = 0; j < 8; ++j)
            pre[j] = ok ? base[(size_t)(2 * j) * SPATIAL] : 0.0f;
    };

    auto storeB = [&](int d, const float* pre) {
        #pragma unroll
        for (int j = 0; j < 8; ++j)
            Bs[d][j * ROWSTRIDE + sb] = pre[j];
    };

    auto compute = [&](int cur) {      // 4 k-steps x 8 WMMAs
        #pragma unroll
        for (int ks = 0; ks < KB / 4; ++ks) {
            const int kpA = 2 * ks + koff2;
            const float* Ab = &As[cur][kpA * ROWSTRIDE + (wm * 32 + row) * 2];
            const float* Bb = &Bs[cur][kpA * ROWSTRIDE + (wn * 64 + row) * 2];
            v2f a0 = *(const v2f*)(Ab);
            v2f a1 = *(const v2f*)(Ab + 32);
            v2f b0 = *(const v2f*)(Bb);
            v2f b1 = *(const v2f*)(Bb + 32);
            v2f b2 = *(const v2f*)(Bb + 64);
            v2f b3 = *(const v2f*)(Bb + 96);
            acc[0][0] = __builtin_amdgcn_wmma_f32_16x16x4_f32(
                false, a0, false, b0, (short)0, acc[0][0], false, false);
            acc[0][1] = __builtin_amdgcn_wmma_f32_16x16x4_f32(
                false, a0, false, b1, (short)0, acc[0][1], false, false);
            acc[0][2] = __builtin_amdgcn_wmma_f32_16x16x4_f32(
                false, a0, false, b2, (short)0, acc[0][2], false, false);
            acc[0][3] = __builtin_amdgcn_wmma_f32_16x16x4_f32(
                false, a0, false, b3, (short)0, acc[0][3], false, false);
            acc[1][0] = __builtin_amdgcn_wmma_f32_16x16x4_f32(
                false, a1, false, b0, (short)0, acc[1][0], false, false);
            acc[1][1] = __builtin_amdgcn_wmma_f32_16x16x4_f32(
                false, a1, false, b1, (short)0, acc[1][1], false, false);
            acc[1][2] = __builtin_amdgcn_wmma_f32_16x16x4_f32(
                false, a1, false, b2, (short)0, acc[1][2], false, false);
            acc[1][3] = __builtin_amdgcn_wmma_f32_16x16x4_f32(
                false, a1, false, b3, (short)0, acc[1][3], false, false);
        }
    };

    // ---- software pipeline over 144 K-chunks, double-buffered LDS ----
    {
        float pre[8];
        stageA(0, 0);
        loadB(0, pre);
        storeB(0, pre);
        wait_async();
    }
    __syncthreads();

    int cur = 0;
    for (int c = 0; c < NCHUNK; ++c) {
        const int nxt = cur ^ 1;
        float pre[8];
        if (c + 1 < NCHUNK) {
            stageA(c + 1, nxt);        // async DMA overlaps compute
            loadB(c + 1, pre);         // global loads in flight over compute
        }
        compute(cur);
        if (c + 1 < NCHUNK) {
            storeB(nxt, pre);          // waits LOADcnt, fills next B panel
            wait_async();              // A panel DMA complete before barrier
        }
        __syncthreads();
        cur = nxt;
    }

    // ---- epilogue: C/D layout (q: M=base+q+(lane>=16?8:0), N=lane&15) ----
    #pragma unroll
    for (int sn = 0; sn < 4; ++sn) {
        int ncol = n0 + wn * 64 + sn * 16 + row;
        int nimg = ncol / SPATIAL;
        int ss   = ncol - nimg * SPATIAL;
        #pragma unroll
        for (int sm = 0; sm < 2; ++sm) {
            int mb = m0 + wm * 32 + sm * 16 + koff2 * 8;
            #pragma unroll
            for (int q = 0; q < 8; ++q) {
                out[((size_t)(nimg * COUT + mb + q)) * SPATIAL + ss] =
                    acc[sm][sn][q];
            }
        }
    }
}

// ---------------------------------------------------------------------------
extern "C" void kernel_launch(void* const* d_in, const int* in_sizes, int n_in,
                              void* d_out, int out_size, void* d_ws, size_t ws_size,
                              hipStream_t stream) {
    const float* x   = (const float*)d_in[0];
    const float* w   = (const float*)d_in[1];
    const float* msk = (const float*)d_in[2];
    float* out = (float*)d_out;
    float* wt  = (float*)d_ws;   // needs 2304*256*4 = 2.36 MB

    prep_weights<<<(KTOT * COUT + 255) / 256, 256, 0, stream>>>(w, msk, wt);

    dim3 grid(COUT / TILE_M, NPIX / TILE_N);   // (2, 784), exact cover
    conv_wmma<<<grid, 256, 0, stream>>>(x, wt, out);
}